// MRConv4d_5016521801767
// MI455X (gfx1250) — compile-verified
//
#include <hip/hip_runtime.h>
#include <hip/hip_bf16.h>
#include <math.h>

typedef float v2f __attribute__((ext_vector_type(2)));
typedef float v8f __attribute__((ext_vector_type(8)));

// Problem constants: B=2, C=32, D=32, H=32, W=64, out_c=32, K(2C)=64
#define VOXB   65536       // voxels per batch (32*32*64)
#define NVOX   131072      // total voxels
#define CHSTR  65536       // channel stride in floats
#define BSTR   2097152     // batch stride in floats (32 ch * 65536)

// -------- gfx1250 async global->LDS copy (ASYNCcnt path), guarded ----------
#if defined(__gfx1250__) && __has_builtin(__builtin_amdgcn_global_load_async_to_lds_b128) && __has_builtin(__builtin_amdgcn_s_wait_asynccnt)
#define HAVE_ASYNC_LDS 1
typedef int v4i_async __attribute__((vector_size(16)));
typedef __attribute__((address_space(1))) v4i_async* gp128_t;   // global int4*
typedef __attribute__((address_space(3))) v4i_async* lp128_t;   // LDS    int4*
#define ASYNC_CP16(gsrc, ldst) \
    __builtin_amdgcn_global_load_async_to_lds_b128((gp128_t)(gsrc), (lp128_t)(ldst), 0, 0)
#define ASYNC_WAIT() __builtin_amdgcn_s_wait_asynccnt(0)
#else
#define HAVE_ASYNC_LDS 0
#endif

// ---------------------------------------------------------------------------
// thresh = mean(norm) - std_unbiased(norm), per batch, from accumulated sums
// ---------------------------------------------------------------------------
__device__ __forceinline__ float compute_thresh(const float* stats, int b) {
    float S1 = stats[b * 2 + 0];
    float S2 = stats[b * 2 + 1];
    const float N = 65536.0f;
    float mean = S1 / N;
    float var  = (S2 - S1 * S1 / N) / (N - 1.0f);   // ddof=1 (torch.std)
    var = fmaxf(var, 0.0f);
    return mean - sqrtf(var);
}

// ---------------------------------------------------------------------------
// Kernel 1: per-voxel L1 norm vs half-rolled x; accumulate sum & sumsq / batch
// roll by (16,16,32) on dims (32,32,64)  ==  p ^ (16*2048 + 16*64 + 32)
// ---------------------------------------------------------------------------
__global__ __launch_bounds__(256) void k_stats(const float* __restrict__ x,
                                               float* __restrict__ stats) {
    int p  = blockIdx.x * 256 + threadIdx.x;   // 0..131071 (grid = 512)
    int b  = p >> 16;
    int pl = p & 65535;
    int p2 = pl ^ 33824;
    size_t base = (size_t)b * BSTR;
    float s = 0.0f;
    #pragma unroll 8
    for (int c = 0; c < 32; ++c)
        s += fabsf(x[base + (size_t)c * CHSTR + pl] - x[base + (size_t)c * CHSTR + p2]);

    __shared__ float r1[256], r2[256];
    int t = threadIdx.x;
    r1[t] = s; r2[t] = s * s;
    __syncthreads();
    for (int k = 128; k > 0; k >>= 1) {
        if (t < k) { r1[t] += r1[t + k]; r2[t] += r2[t + k]; }
        __syncthreads();
    }
    if (t == 0) {   // whole block belongs to one batch (256 | 65536)
        atomicAdd(&stats[b * 2 + 0], r1[0]);
        atomicAdd(&stats[b * 2 + 1], r2[0]);
    }
}

// ---------------------------------------------------------------------------
// Kernel 2: W-axis shifts (s = 0,2,...,62). Block = one (b,d,h) row.
// Writes xj from scratch (max includes 0 via s=0 term).
// ---------------------------------------------------------------------------
__global__ __launch_bounds__(256) void k_axis_w(const float* __restrict__ x,
                                                float* __restrict__ xj,
                                                const float* __restrict__ stats) {
    __shared__ float xs[32 * 64];     // [c][w]
    __shared__ float dist[32 * 64];   // [j][w]
    int blk = blockIdx.x;             // 2048 = b(2)*d(32)*h(32)
    int b = blk >> 10;
    int d = (blk >> 5) & 31;
    int h = blk & 31;
    size_t base = (size_t)b * BSTR + (size_t)d * 2048 + h * 64;
    int t = threadIdx.x;

#if HAVE_ASYNC_LDS
    // 2048 floats = 512 16B copies; rows are 256B contiguous, 16B aligned.
    for (int e4 = t; e4 < 512; e4 += 256) {
        int c = e4 >> 4, w4 = e4 & 15;
        ASYNC_CP16(x + base + (size_t)c * CHSTR + w4 * 4, &xs[e4 * 4]);
    }
    float thr = compute_thresh(stats, b);
    ASYNC_WAIT();
#else
    for (int e = t; e < 2048; e += 256) {
        int c = e >> 6, w = e & 63;
        xs[e] = x[base + (size_t)c * CHSTR + w];
    }
    float thr = compute_thresh(stats, b);
#endif
    __syncthreads();

    // phase 1: dist(w, shift j) = sum_c |x[w] - x[(w-2j)&63]|
    {
        int w  = t & 63;
        int j0 = (t >> 6) * 8;
        for (int jj = 0; jj < 8; ++jj) {
            int j  = j0 + jj;
            int w2 = (w - 2 * j) & 63;
            float s = 0.0f;
            #pragma unroll 8
            for (int c = 0; c < 32; ++c)
                s += fabsf(xs[c * 64 + w] - xs[c * 64 + w2]);
            dist[j * 64 + w] = s;
        }
    }
    __syncthreads();

    // phase 2: thread owns (w, 8 channels): xj = max_j mask*(xr - x)
    {
        int w  = t & 63;
        int c0 = (t >> 6) * 8;
        float xp[8], xr[8];
        #pragma unroll
        for (int k = 0; k < 8; ++k) { xp[k] = xs[(c0 + k) * 64 + w]; xr[k] = 0.0f; }
        for (int j = 0; j < 32; ++j) {
            if (dist[j * 64 + w] < thr) {
                int w2 = (w - 2 * j) & 63;
                #pragma unroll
                for (int k = 0; k < 8; ++k)
                    xr[k] = fmaxf(xr[k], xs[(c0 + k) * 64 + w2] - xp[k]);
            }
        }
        #pragma unroll
        for (int k = 0; k < 8; ++k)
            xj[base + (size_t)(c0 + k) * CHSTR + w] = xr[k];
    }
}

// ---------------------------------------------------------------------------
// Kernel 3: strided-axis shifts (D: NSTRIDE=2048,OSTRIDE=64 / H: 64,2048).
// Tile [c=32][n=32][wi=8] in LDS (32 KB); thread owns one voxel, all channels.
// Read-modify-max into xj.
// ---------------------------------------------------------------------------
template <int NSTRIDE, int OSTRIDE>
__global__ __launch_bounds__(256) void k_axis_n(const float* __restrict__ x,
                                                float* __restrict__ xj,
                                                const float* __restrict__ stats) {
    __shared__ float xs[32 * 32 * 8];   // [c][n][wi]
    int blk = blockIdx.x;               // 512 = b(2)*fo(32)*wc(8)
    int b  = blk >> 8;
    int fo = (blk >> 3) & 31;
    int wc = blk & 7;
    size_t base = (size_t)b * BSTR + (size_t)fo * OSTRIDE + wc * 8;
    int t = threadIdx.x;

#if HAVE_ASYNC_LDS
    // 8192 floats = 2048 16B copies; each (c,n) run is 32B contiguous, aligned.
    for (int e4 = t; e4 < 2048; e4 += 256) {
        int wi4 = e4 & 1, n = (e4 >> 1) & 31, c = e4 >> 6;
        ASYNC_CP16(x + base + (size_t)c * CHSTR + (size_t)n * NSTRIDE + wi4 * 4,
                   &xs[e4 * 4]);
    }
    float thr = compute_thresh(stats, b);
    ASYNC_WAIT();
#else
    for (int e = t; e < 8192; e += 256) {
        int wi = e & 7, n = (e >> 3) & 31, c = e >> 8;
        xs[e] = x[base + (size_t)c * CHSTR + (size_t)n * NSTRIDE + wi];
    }
    float thr = compute_thresh(stats, b);
#endif
    __syncthreads();

    int wi = t & 7;
    int n  = t >> 3;
    float xp[32], xjr[32];
    #pragma unroll
    for (int c = 0; c < 32; ++c) {
        xp[c]  = xs[c * 256 + n * 8 + wi];
        xjr[c] = 0.0f;
    }
    for (int j = 0; j < 16; ++j) {
        int n2 = (n - 2 * j) & 31;
        float xq[32];
        float dist = 0.0f;
        #pragma unroll
        for (int c = 0; c < 32; ++c) {
            xq[c] = xs[c * 256 + n2 * 8 + wi];
            dist += fabsf(xq[c] - xp[c]);
        }
        if (dist < thr) {
            #pragma unroll
            for (int c = 0; c < 32; ++c)
                xjr[c] = fmaxf(xjr[c], xq[c] - xp[c]);
        }
    }
    #pragma unroll
    for (int c = 0; c < 32; ++c) {
        size_t gi = base + (size_t)c * CHSTR + (size_t)n * NSTRIDE + wi;
        xj[gi] = fmaxf(xj[gi], xjr[c]);
    }
}

// ---------------------------------------------------------------------------
// Kernel 4: 1x1x1 conv via V_WMMA_F32_16X16X4_F32.
// y^T[o][v] = sum_c w[o][c] * xcat[c][v] + b[o]   (xcat = [x ; xj], 64 ch)
// Per 16-voxel tile: 2 M-tiles (out 0-15,16-31) x 16 K-steps (K=4 each).
// A frag (16x4): lanes 0-15 -> M=lane, K={k,k+1}; lanes 16-31 -> K={k+2,k+3}.
// B frag (4x16): lanes 0-15 -> N=lane, K={k,k+1}; lanes 16-31 -> K={k+2,k+3}.
// D (16x16): VGPR r -> M = r + 8*hi, N = laneN.
// ---------------------------------------------------------------------------
__global__ __launch_bounds__(256) void k_conv(const float* __restrict__ x,
                                              const float* __restrict__ xj,
                                              const float* __restrict__ w,
                                              const float* __restrict__ bias,
                                              float* __restrict__ y) {
    int tid   = blockIdx.x * blockDim.x + threadIdx.x;
    int wave  = tid >> 5;
    int lane  = threadIdx.x & 31;
    int laneN = lane & 15;
    int hi    = lane >> 4;
    int nwaves = (gridDim.x * blockDim.x) >> 5;

    // Preload A fragments (w is [32 out][64 ch] row-major)
    v2f a[2][16];
    #pragma unroll
    for (int m = 0; m < 2; ++m) {
        #pragma unroll
        for (int k = 0; k < 16; ++k) {
            int row = m * 16 + laneN;
            int col = k * 4 + 2 * hi;
            a[m][k].x = w[row * 64 + col];
            a[m][k].y = w[row * 64 + col + 1];
        }
    }
    // Tile-invariant bias per D-layout lane slot
    float b0[8], b1[8];
    #pragma unroll
    for (int r = 0; r < 8; ++r) {
        b0[r] = bias[r + hi * 8];
        b1[r] = bias[16 + r + hi * 8];
    }

    const int NTILE = NVOX / 16;   // 8192
    for (int tile = wave; tile < NTILE; tile += nwaves) {
        int v0 = tile * 16;
        int b  = v0 >> 16;
        size_t xbase = (size_t)b * BSTR + (v0 & 65535) + laneN;

        // prefetch a future tile's data (lowers to global_prefetch_b8)
        {
            int tn = tile + nwaves;
            if (tn < NTILE) {
                int vn = tn * 16;
                size_t nb = (size_t)(vn >> 16) * BSTR + (vn & 65535) + laneN;
                __builtin_prefetch(x + nb, 0, 0);
                __builtin_prefetch(xj + nb, 0, 0);
            }
        }

        v8f acc0 = {}, acc1 = {};
        #pragma unroll
        for (int k = 0; k < 16; ++k) {
            int c0 = k * 4 + 2 * hi;            // even; pair never straddles 32
            const float* src = (c0 < 32) ? (x + xbase + (size_t)c0 * CHSTR)
                                         : (xj + xbase + (size_t)(c0 - 32) * CHSTR);
            v2f bf;
            bf.x = src[0];
            bf.y = src[CHSTR];
            acc0 = __builtin_amdgcn_wmma_f32_16x16x4_f32(
                false, a[0][k], false, bf, (short)0, acc0, false, false);
            acc1 = __builtin_amdgcn_wmma_f32_16x16x4_f32(
                false, a[1][k], false, bf, (short)0, acc1, false, false);
        }

        #pragma unroll
        for (int r = 0; r < 8; ++r) {
            int o0 = r + hi * 8;
            int o1 = 16 + o0;
            y[(size_t)o0 * NVOX + v0 + laneN] = acc0[r] + b0[r];
            y[(size_t)o1 * NVOX + v0 + laneN] = acc1[r] + b1[r];
        }
    }
}

// ---------------------------------------------------------------------------
// Kernel 5: per-out-channel sums of y and y^2 (one block per channel)
// ---------------------------------------------------------------------------
__global__ __launch_bounds__(256) void k_bnstats(const float* __restrict__ y,
                                                 float* __restrict__ s1,
                                                 float* __restrict__ s2) {
    int o = blockIdx.x;   // 32 blocks
    const float* yo = y + (size_t)o * NVOX;
    float a1 = 0.0f, a2 = 0.0f;
    for (int i = threadIdx.x; i < NVOX; i += 256) {
        float v = yo[i];
        a1 += v; a2 += v * v;
    }
    __shared__ float r1[256], r2[256];
    int t = threadIdx.x;
    r1[t] = a1; r2[t] = a2;
    __syncthreads();
    for (int k = 128; k > 0; k >>= 1) {
        if (t < k) { r1[t] += r1[t + k]; r2[t] += r2[t + k]; }
        __syncthreads();
    }
    if (t == 0) { s1[o] = r1[0]; s2[o] = r2[0]; }
}

// ---------------------------------------------------------------------------
// Kernel 6: BN (batch stats, biased var) + exact GELU, write [B,O,D,H,W]
// ---------------------------------------------------------------------------
__global__ __launch_bounds__(256) void k_bngelu(const float* __restrict__ y,
                                                const float* __restrict__ gamma,
                                                const float* __restrict__ beta,
                                                const float* __restrict__ s1,
                                                const float* __restrict__ s2,
                                                float* __restrict__ out) {
    int i = blockIdx.x * 256 + threadIdx.x;   // 0..4194303
    int b = i >> 21;
    int o = (i >> 16) & 31;
    int p = i & 65535;
    const float invN = 1.0f / (float)NVOX;
    float mu  = s1[o] * invN;
    float var = s2[o] * invN - mu * mu;
    float rs  = rsqrtf(var + 1e-5f);
    float v   = y[(size_t)o * NVOX + (b << 16) + p];
    float z   = gamma[o] * (v - mu) * rs + beta[o];
    out[i] = 0.5f * z * (1.0f + erff(z * 0.70710678118654752f));
}

// ---------------------------------------------------------------------------
extern "C" void kernel_launch(void* const* d_in, const int* in_sizes, int n_in,
                              void* d_out, int out_size, void* d_ws, size_t ws_size,
                              hipStream_t stream) {
    (void)in_sizes; (void)n_in; (void)out_size; (void)ws_size;
    const float* x     = (const float*)d_in[0];
    const float* w     = (const float*)d_in[1];
    const float* bias  = (const float*)d_in[2];
    const float* gamma = (const float*)d_in[3];
    const float* beta  = (const float*)d_in[4];
    float* out = (float*)d_out;

    float* ws    = (float*)d_ws;
    float* xj    = ws;                      // 4,194,304 floats
    float* y     = ws + 4194304;            // 4,194,304 floats
    float* stats = ws + 8388608;            // 4 floats (S1,S2 per batch)
    float* bns1  = stats + 4;               // 32 floats
    float* bns2  = bns1 + 32;               // 32 floats

    (void)hipMemsetAsync(stats, 0, 4 * sizeof(float), stream);

    k_stats  <<<512,  256, 0, stream>>>(x, stats);
    k_axis_w <<<2048, 256, 0, stream>>>(x, xj, stats);      // W shifts (writes xj)
    k_axis_n<64,   2048><<<512, 256, 0, stream>>>(x, xj, stats);  // H shifts
    k_axis_n<2048, 64  ><<<512, 256, 0, stream>>>(x, xj, stats);  // D shifts
    k_conv   <<<64,   256, 0, stream>>>(x, xj, w, bias, y);
    k_bnstats<<<32,   256, 0, stream>>>(y, bns1, bns2);
    k_bngelu <<<16384,256, 0, stream>>>(y, gamma, beta, bns1, bns2, out);
}